// GCN_39556648796618
// MI455X (gfx1250) — compile-verified
//
#include <hip/hip_runtime.h>
#include <hip/hip_bf16.h>
#include <math.h>

typedef __attribute__((ext_vector_type(2))) float v2f;
typedef __attribute__((ext_vector_type(8))) float v8f;

#define N_NODES  60000
#define N_EDGES  960000
#define N_GRAPHS 64
#define F_IN     64
#define F_H      128
#define F_OUT_   64

// ---------------- degree / normalization ----------------
__global__ void init_deg(float* __restrict__ deg) {
    int i = blockIdx.x * blockDim.x + threadIdx.x;
    if (i < N_NODES) deg[i] = 1.0f;              // self-loop contribution
}

__global__ void accum_deg(const int* __restrict__ dst, float* __restrict__ deg) {
    int e = blockIdx.x * blockDim.x + threadIdx.x;
    if (e < N_EDGES) atomicAdd(&deg[dst[e]], 1.0f);
}

__global__ void deg_to_dinv(float* __restrict__ deg) {
    int i = blockIdx.x * blockDim.x + threadIdx.x;
    if (i < N_NODES) {
        float d = deg[i];
        deg[i] = d > 0.0f ? rsqrtf(d) : 0.0f;
    }
}

// ---------------- dense GEMM via fp32 WMMA ----------------
// out[N_NODES x NC] = act(A[N_NODES x K]) @ W[K x NC]
// One wave computes one 16x16 tile; B fragments for the whole K strip are
// preloaded into registers (W is tiny -> L2/WGP$ resident).
template<int K, int NC, bool RELU>
__global__ void __launch_bounds__(256)
gemm_wmma(const float* __restrict__ A, const float* __restrict__ W,
          float* __restrict__ out) {
    const int lane  = threadIdx.x & 31;
    const int wave  = threadIdx.x >> 5;
    const int ct    = blockIdx.y;                    // 16-col tile of output
    const int nRowTiles  = N_NODES / 16;             // 3750
    const int wavesTotal = gridDim.x * 8;
    const int waveId     = blockIdx.x * 8 + wave;

    const int mLane = lane & 15;                     // M (A) / N (B,C) index
    const int kHalf = (lane >> 4) << 1;              // 0 or 2

    // Preload B fragments: lane holds W[kb][n], W[kb+1][n]
    v2f bfrag[K / 4];
    const int n = ct * 16 + mLane;
#pragma unroll
    for (int j = 0; j < K / 4; ++j) {
        int kb = 4 * j + kHalf;
        v2f b;
        b.x = W[kb * NC + n];
        b.y = W[(kb + 1) * NC + n];
        bfrag[j] = b;
    }

    for (int rt = waveId; rt < nRowTiles; rt += wavesTotal) {
        v8f c = {0.f, 0.f, 0.f, 0.f, 0.f, 0.f, 0.f, 0.f};
        const float* arow = A + (size_t)(rt * 16 + mLane) * K;
#pragma unroll
        for (int j = 0; j < K / 4; ++j) {
            int kb = 4 * j + kHalf;                  // even -> 8B aligned
            v2f a = *(const v2f*)(arow + kb);        // one b64 load
            if (RELU) { a.x = fmaxf(a.x, 0.f); a.y = fmaxf(a.y, 0.f); }
            c = __builtin_amdgcn_wmma_f32_16x16x4_f32(
                    /*neg_a=*/false, a, /*neg_b=*/false, bfrag[j],
                    /*c_mod=*/(short)0, c, /*reuse_a=*/false, /*reuse_b=*/false);
        }
        float* orow = out + (size_t)rt * 16 * NC + ct * 16;
        const int rbase = (lane >> 4) << 3;          // 0 or 8
#pragma unroll
        for (int g = 0; g < 8; ++g)
            orow[(size_t)(rbase + g) * NC + mLane] = c[g];
    }
}

// ---------------- self-loop + bias init ----------------
template<int F>
__global__ void self_bias(const float* __restrict__ hlin,
                          const float* __restrict__ dinv,
                          const float* __restrict__ bias,
                          float* __restrict__ out) {
    int idx = blockIdx.x * blockDim.x + threadIdx.x;
    if (idx >= N_NODES * F) return;
    int node = idx / F;
    int f    = idx % F;
    float di = dinv[node];
    out[idx] = bias[f] + di * di * hlin[idx];
}

// ---------------- edge aggregation (one wave per edge) ----------------
template<int F>
__global__ void agg_edges(const float* __restrict__ hlin,
                          const int* __restrict__ src,
                          const int* __restrict__ dst,
                          const float* __restrict__ dinv,
                          float* __restrict__ out) {
    int gw   = (blockIdx.x * blockDim.x + threadIdx.x) >> 5;
    int lane = threadIdx.x & 31;
    if (gw >= N_EDGES) return;
    int s = src[gw];
    int d = dst[gw];
    float w = dinv[s] * dinv[d];
    constexpr int PER = F / 32;                      // 4 (F=128) or 2 (F=64)
    const float* hp = hlin + (size_t)s * F + lane * PER;
    float*       op = out  + (size_t)d * F + lane * PER;
    float v[PER];
#pragma unroll
    for (int i = 0; i < PER; ++i) v[i] = hp[i];      // coalesced gather
#pragma unroll
    for (int i = 0; i < PER; ++i) atomicAdd(op + i, w * v[i]);
}

// ---------------- pooling + FC head ----------------
__global__ void zero_pool(float* __restrict__ pooled, float* __restrict__ counts) {
    int i = blockIdx.x * blockDim.x + threadIdx.x;
    if (i < N_GRAPHS * F_OUT_) pooled[i] = 0.0f;
    if (i < N_GRAPHS)          counts[i] = 0.0f;
}

__global__ void count_nodes(const int* __restrict__ batch, float* __restrict__ counts) {
    int i = blockIdx.x * blockDim.x + threadIdx.x;
    if (i < N_NODES) atomicAdd(&counts[batch[i]], 1.0f);
}

__global__ void pool_sum(const float* __restrict__ h, const int* __restrict__ batch,
                         float* __restrict__ pooled) {
    int idx = blockIdx.x * blockDim.x + threadIdx.x;
    if (idx >= N_NODES * F_OUT_) return;
    int node = idx / F_OUT_;
    int f    = idx & (F_OUT_ - 1);
    float v  = fmaxf(h[idx], 0.0f);                  // final ReLU fused here
    atomicAdd(&pooled[batch[node] * F_OUT_ + f], v);
}

__global__ void final_fc(const float* __restrict__ pooled,
                         const float* __restrict__ counts,
                         const float* __restrict__ Wfc,
                         const float* __restrict__ bfc,
                         float* __restrict__ out) {
    int g = threadIdx.x;
    if (g >= N_GRAPHS) return;
    float sum = 0.0f;
    for (int f = 0; f < F_OUT_; ++f) sum += pooled[g * F_OUT_ + f] * Wfc[f];
    float cnt = fmaxf(counts[g], 1.0f);
    float z   = sum / cnt + bfc[0];
    out[g] = 1.0f / (1.0f + expf(-z));
}

// ---------------- host launcher ----------------
extern "C" void kernel_launch(void* const* d_in, const int* in_sizes, int n_in,
                              void* d_out, int out_size, void* d_ws, size_t ws_size,
                              hipStream_t stream) {
    (void)in_sizes; (void)n_in; (void)out_size; (void)ws_size;

    const float* x     = (const float*)d_in[0];
    const int*   ei    = (const int*)  d_in[1];
    const int*   batch = (const int*)  d_in[2];
    const float* W1    = (const float*)d_in[3];
    const float* b1    = (const float*)d_in[4];
    const float* W2    = (const float*)d_in[5];
    const float* b2    = (const float*)d_in[6];
    const float* W3    = (const float*)d_in[7];
    const float* b3    = (const float*)d_in[8];
    const float* Wfc   = (const float*)d_in[9];
    const float* bfc   = (const float*)d_in[10];
    float* out = (float*)d_out;

    const int* src = ei;             // edge_index[0]
    const int* dst = ei + N_EDGES;   // edge_index[1]

    float* ws     = (float*)d_ws;
    float* dinv   = ws;  ws += N_NODES;
    float* counts = ws;  ws += N_GRAPHS;
    float* pooled = ws;  ws += N_GRAPHS * F_OUT_;
    float* bufA   = ws;  ws += (size_t)N_NODES * F_H;
    float* bufB   = ws;  ws += (size_t)N_NODES * F_H;

    const int T = 256;
    auto cdiv = [](long long a, long long b) { return (int)((a + b - 1) / b); };

    // norm = dinv[src]*dinv[dst], dinv = rsqrt(indeg+1)
    init_deg  <<<cdiv(N_NODES, T), T, 0, stream>>>(dinv);
    accum_deg <<<cdiv(N_EDGES, T), T, 0, stream>>>(dst, dinv);
    deg_to_dinv<<<cdiv(N_NODES, T), T, 0, stream>>>(dinv);

    const int gemmBlocks = (N_NODES / 16 + 7) / 8;     // 8 waves per block
    const int edgeBlocks = cdiv((long long)N_EDGES * 32, T);

    // Layer 1: lin = x@W1 -> bufA ; out1 = b1 + self + edges -> bufB
    gemm_wmma<F_IN, F_H, false><<<dim3(gemmBlocks, F_H / 16), T, 0, stream>>>(x, W1, bufA);
    self_bias<F_H><<<cdiv((long long)N_NODES * F_H, T), T, 0, stream>>>(bufA, dinv, b1, bufB);
    agg_edges<F_H><<<edgeBlocks, T, 0, stream>>>(bufA, src, dst, dinv, bufB);

    // Layer 2: lin = relu(bufB)@W2 -> bufA ; out2 -> bufB
    gemm_wmma<F_H, F_H, true><<<dim3(gemmBlocks, F_H / 16), T, 0, stream>>>(bufB, W2, bufA);
    self_bias<F_H><<<cdiv((long long)N_NODES * F_H, T), T, 0, stream>>>(bufA, dinv, b2, bufB);
    agg_edges<F_H><<<edgeBlocks, T, 0, stream>>>(bufA, src, dst, dinv, bufB);

    // Layer 3: lin = relu(bufB)@W3 -> bufA (N x 64) ; out3 -> bufB (N x 64)
    gemm_wmma<F_H, F_OUT_, true><<<dim3(gemmBlocks, F_OUT_ / 16), T, 0, stream>>>(bufB, W3, bufA);
    self_bias<F_OUT_><<<cdiv((long long)N_NODES * F_OUT_, T), T, 0, stream>>>(bufA, dinv, b3, bufB);
    agg_edges<F_OUT_><<<edgeBlocks, T, 0, stream>>>(bufA, src, dst, dinv, bufB);

    // Mean-pool per graph + FC + sigmoid
    zero_pool  <<<cdiv(N_GRAPHS * F_OUT_, T), T, 0, stream>>>(pooled, counts);
    count_nodes<<<cdiv(N_NODES, T), T, 0, stream>>>(batch, counts);
    pool_sum   <<<cdiv((long long)N_NODES * F_OUT_, T), T, 0, stream>>>(bufB, batch, pooled);
    final_fc   <<<1, 64, 0, stream>>>(pooled, counts, Wfc, bfc, out);
}